// HaarDWT_85899346022
// MI455X (gfx1250) — compile-verified
//
#include <hip/hip_runtime.h>

typedef float v4f __attribute__((ext_vector_type(4)));

// Geometry from the reference: x is (8, 64, 512, 512) fp32.
#define IN_W        512
#define IN_H        512
#define OUT_W       256
#define OUT_H       256
#define OUT_PLANE   (OUT_W * OUT_W)              // 65536 floats per output plane
#define TILE_OROWS  4                            // output rows per tile
#define TILE_BYTES  (TILE_OROWS * 2 * IN_W * 4)  // 8 input rows * 2KB = 16384 B (contiguous)
#define TILE_FLOATS (TILE_BYTES / 4)             // 4096 floats
#define THREADS     256                          // 8 wave32s
#define NBUF        4                            // 4-deep LDS rotation, 64 KiB total

// Async-copy one contiguous 16 KiB input tile into LDS.
// Each of the 256 threads issues 4x GLOBAL_LOAD_ASYNC_TO_LDS_B128 (ASYNCcnt += 4/wave).
// Input is streamed exactly once (512 MiB > 192 MB L2) -> non-temporal load policy.
__device__ __forceinline__ void async_copy_tile(const float* __restrict__ x,
                                                int tile, unsigned lds_base, int t) {
  const char* gp = (const char*)x + (size_t)tile * TILE_BYTES;
#pragma unroll
  for (int j = 0; j < 4; ++j) {
    unsigned lofs = lds_base + (unsigned)(j * 4096 + t * 16);
    const char* ga = gp + j * 4096 + t * 16;
    asm volatile("global_load_async_to_lds_b128 %0, %1, off th:TH_LOAD_NT"
                 :: "v"(lofs), "v"(ga)
                 : "memory");
  }
}

__global__ void __launch_bounds__(THREADS)
haar_dwt_kernel(const float* __restrict__ x, float* __restrict__ out, int num_tiles) {
  __shared__ float sbuf[NBUF][TILE_FLOATS];      // 64 KiB, 4-slot rotation
  const int t = threadIdx.x;
  // Low 32 bits of a flat LDS pointer are the LDS byte offset (ISA 10.2 aperture rule).
  unsigned lb[NBUF];
#pragma unroll
  for (int i = 0; i < NBUF; ++i) lb[i] = (unsigned)(unsigned long long)&sbuf[i][0];

  const size_t out_tensor = (size_t)(num_tiles >> 6) * OUT_PLANE;  // planes * 65536
  float* __restrict__ out_ll = out;
  float* __restrict__ out_lh = out + out_tensor;
  float* __restrict__ out_hl = out + 2 * out_tensor;
  float* __restrict__ out_hh = out + 3 * out_tensor;

  const int g = (int)gridDim.x;
  int ti = blockIdx.x;

  // Prime the pipeline two tiles deep.
  if (ti < num_tiles)     async_copy_tile(x, ti,     lb[0], t);
  if (ti + g < num_tiles) async_copy_tile(x, ti + g, lb[1], t);

  int k = 0;
  for (; ti < num_tiles; ti += g, k = (k + 1) & (NBUF - 1)) {
    const bool n1 = (ti + g)     < num_tiles;
    const bool n2 = (ti + 2 * g) < num_tiles;

    // Keep two tiles in flight: slot (k+2)&3 was last READ at iteration i-2,
    // which every wave finished before iteration i-1's barrier -> safe to
    // overwrite without a trailing barrier.
    if (n2) {
      async_copy_tile(x, ti + 2 * g, lb[(k + 2) & (NBUF - 1)], t);
      asm volatile("s_wait_asynccnt 0x8" ::: "memory");   // current tile resident
    } else if (n1) {
      asm volatile("s_wait_asynccnt 0x4" ::: "memory");
    } else {
      asm volatile("s_wait_asynccnt 0x0" ::: "memory");
    }
    __syncthreads();   // single barrier per tile: all waves' copies visible

    const float* __restrict__ buf = sbuf[k];
    const int r  = t >> 6;      // output row within tile (0..3)
    const int cg = t & 63;      // column group: 4 output px -> 8 input floats/row

    const float* e = buf + (2 * r) * IN_W + cg * 8;    // even input row
    v4f ev0 = *(const v4f*)(e);                        // a0 b0 a1 b1
    v4f ev1 = *(const v4f*)(e + 4);                    // a2 b2 a3 b3
    v4f od0 = *(const v4f*)(e + IN_W);                 // c0 d0 c1 d1
    v4f od1 = *(const v4f*)(e + IN_W + 4);             // c2 d2 c3 d3

    v4f s1 = { ev0.x + ev0.y, ev0.z + ev0.w, ev1.x + ev1.y, ev1.z + ev1.w }; // a+b
    v4f d1 = { ev0.y - ev0.x, ev0.w - ev0.z, ev1.y - ev1.x, ev1.w - ev1.z }; // b-a
    v4f s2 = { od0.x + od0.y, od0.z + od0.w, od1.x + od1.y, od1.z + od1.w }; // c+d
    v4f d2 = { od0.y - od0.x, od0.w - od0.z, od1.y - od1.x, od1.w - od1.z }; // d-c

    v4f ll = (s1 + s2) * 0.5f;   // ( a+b+c+d)/2
    v4f lh = (s2 - s1) * 0.5f;   // (-a-b+c+d)/2
    v4f hl = (d1 + d2) * 0.5f;   // (-a+b-c+d)/2
    v4f hh = (d2 - d1) * 0.5f;   // ( a-b-c+d)/2

    const int p  = ti >> 6;      // plane (64 tiles per plane)
    const int tr = ti & 63;      // tile row within plane
    const size_t ob = (size_t)p * OUT_PLANE
                    + (size_t)(tr * TILE_OROWS + r) * OUT_W
                    + (size_t)cg * 4;

    // Streaming outputs: non-temporal B128 stores (don't thrash 192MB L2).
    __builtin_nontemporal_store(ll, (v4f*)(out_ll + ob));
    __builtin_nontemporal_store(lh, (v4f*)(out_lh + ob));
    __builtin_nontemporal_store(hl, (v4f*)(out_hl + ob));
    __builtin_nontemporal_store(hh, (v4f*)(out_hh + ob));
  }
}

extern "C" void kernel_launch(void* const* d_in, const int* in_sizes, int n_in,
                              void* d_out, int out_size, void* d_ws, size_t ws_size,
                              hipStream_t stream) {
  const float* x = (const float*)d_in[0];
  float* out = (float*)d_out;

  const int planes    = in_sizes[0] / (IN_W * IN_H);        // 8*64 = 512
  const int num_tiles = planes * (OUT_H / TILE_OROWS);      // 512*64 = 32768

  int grid = 4096;                                          // persistent blocks, ~8 tiles each
  if (grid > num_tiles) grid = num_tiles;
  haar_dwt_kernel<<<grid, THREADS, 0, stream>>>(x, out, num_tiles);
}